// GCNEncoder_85126251807435
// MI455X (gfx1250) — compile-verified
//
#include <hip/hip_runtime.h>

#define N_NODES 50000
#define N_EDGES 1600000
#define F_IN    128
#define F_HID   64
#define F_OUT   32

typedef __attribute__((ext_vector_type(2))) float v2f;
typedef __attribute__((ext_vector_type(8))) float v8f;

// ---------------- utility ----------------

__global__ void zero_kernel(float* __restrict__ p, unsigned n) {
    unsigned i = blockIdx.x * blockDim.x + threadIdx.x;
    if (i < n) p[i] = 0.0f;
}

__global__ void deg_kernel(const int* __restrict__ ei, float* __restrict__ deg) {
    unsigned e = blockIdx.x * blockDim.x + threadIdx.x;
    if (e < N_EDGES) {
        int c = ei[N_EDGES + e];          // col = target
        atomicAdd(&deg[c], 1.0f);
    }
}

__global__ void dinv_kernel(float* __restrict__ deg) {
    unsigned i = blockIdx.x * blockDim.x + threadIdx.x;
    if (i < N_NODES) deg[i] = rsqrtf(deg[i] + 1.0f);
}

// ---------------- GEMM1: H[50000x64] = X[50000x128] @ W1[128x64], f32 WMMA ----------------
// Block = 128 threads = 4 waves; wave w computes the 16x16 tile at cols [16w,16w+16).
// Grid = 50000/16 = 3125 blocks exactly (no tail -> EXEC all ones for WMMA).

__global__ __launch_bounds__(128) void gemm1_wmma(const float* __restrict__ X,
                                                  const float* __restrict__ W,
                                                  float* __restrict__ H) {
    const int lane    = threadIdx.x & 31;
    const int wave    = threadIdx.x >> 5;
    const int l       = lane & 15;
    const int half    = lane >> 4;              // 0: K pair {0,1}; 1: K pair {2,3}
    const int rowTile = blockIdx.x << 4;
    const int colTile = wave << 4;

    const float* xrow = X + (size_t)(rowTile + l) * F_IN + 2 * half;       // A: lane = row M
    const float* wcol = W + (size_t)(2 * half) * F_HID + colTile + l;      // B: lane = col N

    v8f c = {};
#pragma unroll 4
    for (int k = 0; k < F_IN; k += 4) {
        v2f a, b;
        a.x = xrow[k];                    // A[M][k + 2*half + 0]
        a.y = xrow[k + 1];                // A[M][k + 2*half + 1]
        b.x = wcol[(size_t)k * F_HID];          // B[k + 2*half + 0][N]
        b.y = wcol[(size_t)(k + 1) * F_HID];    // B[k + 2*half + 1][N]
        c = __builtin_amdgcn_wmma_f32_16x16x4_f32(false, a, false, b,
                                                  (short)0, c, false, false);
    }

    // C/D layout: VGPR i -> M = i + 8*half, N = lane&15
    float* hrow = H + (size_t)(rowTile + 8 * half) * F_HID + colTile + l;
#pragma unroll
    for (int i = 0; i < 8; ++i) hrow[(size_t)i * F_HID] = c[i];
}

// ---------------- GEMM2: H2(mu|ls)[2][50000x32] = H1[50000x64] @ {Wmu,Wls}[64x32] ----------------
// 4 waves/block: wave0/1 -> mu cols 0-15 / 16-31; wave2/3 -> ls cols 0-15 / 16-31.

__global__ __launch_bounds__(128) void gemm2_wmma(const float* __restrict__ H1,
                                                  const float* __restrict__ Wmu,
                                                  const float* __restrict__ Wls,
                                                  float* __restrict__ H2) {
    const int lane    = threadIdx.x & 31;
    const int wave    = threadIdx.x >> 5;
    const int l       = lane & 15;
    const int half    = lane >> 4;
    const int rowTile = blockIdx.x << 4;
    const int colTile = (wave & 1) << 4;
    const int plane   = wave >> 1;              // 0=mu, 1=logstd

    const float* W   = plane ? Wls : Wmu;
    float*       out = H2 + (size_t)plane * ((size_t)N_NODES * F_OUT);

    const float* xrow = H1 + (size_t)(rowTile + l) * F_HID + 2 * half;
    const float* wcol = W + (size_t)(2 * half) * F_OUT + colTile + l;

    v8f c = {};
#pragma unroll 4
    for (int k = 0; k < F_HID; k += 4) {
        v2f a, b;
        a.x = xrow[k];
        a.y = xrow[k + 1];
        b.x = wcol[(size_t)k * F_OUT];
        b.y = wcol[(size_t)(k + 1) * F_OUT];
        c = __builtin_amdgcn_wmma_f32_16x16x4_f32(false, a, false, b,
                                                  (short)0, c, false, false);
    }

    float* orow = out + (size_t)(rowTile + 8 * half) * F_OUT + colTile + l;
#pragma unroll
    for (int i = 0; i < 8; ++i) orow[(size_t)i * F_OUT] = c[i];
}

// ---------------- edge scatter, layer 1: agg[col] += h[row] * dinv[row]*dinv[col] ----------------
// One thread per (edge, 4-feature chunk): float4 gather + 4 f32 atomics (L2-resident).

__global__ void scatter1(const int* __restrict__ ei, const float* __restrict__ dinv,
                         const float* __restrict__ H, float* __restrict__ agg) {
    unsigned idx = blockIdx.x * blockDim.x + threadIdx.x;
    if (idx >= (unsigned)N_EDGES * (F_HID / 4)) return;   // 25.6M
    unsigned e  = idx >> 4;                                // /16 chunks per edge
    unsigned f4 = (idx & 15) << 2;
    int r = ei[e];
    int t = ei[N_EDGES + e];
    float norm = dinv[r] * dinv[t];
    float4 v = *(const float4*)(H + (size_t)r * F_HID + f4);
    float* dst = agg + (size_t)t * F_HID + f4;
    atomicAdd(dst + 0, v.x * norm);
    atomicAdd(dst + 1, v.y * norm);
    atomicAdd(dst + 2, v.z * norm);
    atomicAdd(dst + 3, v.w * norm);
}

// h1 = relu(agg + h*dinv^2 + b1), in place in agg
__global__ void finalize1(const float* __restrict__ H, const float* __restrict__ dinv,
                          const float* __restrict__ b1, float* __restrict__ agg) {
    unsigned idx = blockIdx.x * blockDim.x + threadIdx.x;
    if (idx >= (unsigned)N_NODES * F_HID) return;
    unsigned n = idx >> 6;
    unsigned f = idx & (F_HID - 1);
    float di = dinv[n];
    float v = agg[idx] + H[idx] * di * di + b1[f];
    agg[idx] = v > 0.0f ? v : 0.0f;
}

// ---------------- edge scatter, layer 2: both planes (mu, logstd) ----------------

__global__ void scatter2(const int* __restrict__ ei, const float* __restrict__ dinv,
                         const float* __restrict__ H2, float* __restrict__ out) {
    unsigned idx = blockIdx.x * blockDim.x + threadIdx.x;
    if (idx >= (unsigned)N_EDGES * (2 * F_OUT / 4)) return;  // 25.6M
    unsigned e  = idx >> 4;
    unsigned q  = idx & 15;            // 8 chunks mu, 8 chunks ls
    unsigned pl = q >> 3;
    unsigned f4 = (q & 7) << 2;
    int r = ei[e];
    int t = ei[N_EDGES + e];
    float norm = dinv[r] * dinv[t];
    size_t base = (size_t)pl * ((size_t)N_NODES * F_OUT);
    float4 v = *(const float4*)(H2 + base + (size_t)r * F_OUT + f4);
    float* dst = out + base + (size_t)t * F_OUT + f4;
    atomicAdd(dst + 0, v.x * norm);
    atomicAdd(dst + 1, v.y * norm);
    atomicAdd(dst + 2, v.z * norm);
    atomicAdd(dst + 3, v.w * norm);
}

// out += h2*dinv^2 + bias  (mu plane then logstd plane)
__global__ void finalize2(const float* __restrict__ H2, const float* __restrict__ dinv,
                          const float* __restrict__ bmu, const float* __restrict__ bls,
                          float* __restrict__ out) {
    unsigned idx = blockIdx.x * blockDim.x + threadIdx.x;
    if (idx >= (unsigned)N_NODES * 2 * F_OUT) return;
    unsigned plane = idx / ((unsigned)N_NODES * F_OUT);
    unsigned rem   = idx - plane * ((unsigned)N_NODES * F_OUT);
    unsigned n = rem >> 5;
    unsigned f = rem & (F_OUT - 1);
    float di = dinv[n];
    const float* b = plane ? bls : bmu;
    out[idx] += H2[idx] * di * di + b[f];
}

// ---------------- launch ----------------

extern "C" void kernel_launch(void* const* d_in, const int* in_sizes, int n_in,
                              void* d_out, int out_size, void* d_ws, size_t ws_size,
                              hipStream_t stream) {
    const float* x   = (const float*)d_in[0];
    const int*   ei  = (const int*)d_in[1];
    const float* W1  = (const float*)d_in[2];
    const float* b1  = (const float*)d_in[3];
    const float* Wmu = (const float*)d_in[4];
    const float* bmu = (const float*)d_in[5];
    const float* Wls = (const float*)d_in[6];
    const float* bls = (const float*)d_in[7];
    float* out = (float*)d_out;

    // workspace layout (floats): dinv | H (reused as H2) | agg1/h1
    float* ws   = (float*)d_ws;
    float* dinv = ws;                                   // N_NODES (padded to 50176)
    float* H    = ws + 50176;                           // N*64  (later reused: H2 = 2 * N*32)
    float* agg  = H + (size_t)N_NODES * F_HID;          // N*64  -> h1 after finalize1

    const unsigned nDeg  = N_NODES;
    const unsigned nAgg  = (unsigned)N_NODES * F_HID;       // 3.2M
    const unsigned nOut  = (unsigned)N_NODES * 2 * F_OUT;   // 3.2M
    const unsigned nSc   = (unsigned)N_EDGES * 16;          // 25.6M chunk threads

    zero_kernel<<<(nDeg + 255) / 256, 256, 0, stream>>>(dinv, nDeg);
    zero_kernel<<<(nAgg + 255) / 256, 256, 0, stream>>>(agg, nAgg);
    zero_kernel<<<(nOut + 255) / 256, 256, 0, stream>>>(out, nOut);

    deg_kernel<<<(N_EDGES + 255) / 256, 256, 0, stream>>>(ei, dinv);
    dinv_kernel<<<(N_NODES + 255) / 256, 256, 0, stream>>>(dinv);

    gemm1_wmma<<<N_NODES / 16, 128, 0, stream>>>(x, W1, H);

    scatter1<<<(nSc + 255) / 256, 256, 0, stream>>>(ei, dinv, H, agg);
    finalize1<<<(nAgg + 255) / 256, 256, 0, stream>>>(H, dinv, b1, agg);

    gemm2_wmma<<<N_NODES / 16, 128, 0, stream>>>(agg, Wmu, Wls, H);

    scatter2<<<(nSc + 255) / 256, 256, 0, stream>>>(ei, dinv, H, out);
    finalize2<<<(nOut + 255) / 256, 256, 0, stream>>>(H, dinv, bmu, bls, out);
}